// EuclideanCodebook_61521111547966
// MI455X (gfx1250) — compile-verified
//
#include <hip/hip_runtime.h>
#include <hip/hip_bf16.h>

// CDNA5 / gfx1250 VQ codebook nearest-neighbor (VQ-VAE EuclideanCodebook).
// Full-f32 matrix path via V_WMMA_F32_16X16X4_F32: bf16/fp8 would corrupt the
// argmin (code-distance gaps ~1e-6 < bf16-induced dot error ~5e-6).
//
// Round 2: inner loop reduced to {ds_load_b64, global_load_b64, v_wmma} by
// (a) exploiting the 16x4-f32 A/B VGPR layouts (each lane's fragment is a
//     contiguous 8B pair at base + 2*khalf) -> no v_cndmask selects, and
// (b) precomputing ||w_k||^2 into d_ws once -> no mul/fmac in the hot loop.

typedef float v2f __attribute__((ext_vector_type(2)));
typedef float v8f __attribute__((ext_vector_type(8)));

#define Bv    16
#define Hd    64
#define Wd    64
#define Cd    512
#define Kd    4096
#define NROW  (Bv * Hd * Wd)      // 65536 query vectors
#define WAVES 4
#define NTHR  (WAVES * 32)
#define KTILE (WAVES * 16)        // 64 codes per k-iteration per block

__global__ __launch_bounds__(NTHR)
void vq_argmin_kernel(const float* __restrict__ x,
                      const float* __restrict__ weight,
                      const float* __restrict__ wsq,
                      float* __restrict__ out_q,
                      float* __restrict__ out_idx,
                      int*   __restrict__ hist)
{
    __shared__ float xs[16 * Cd];       // 32 KB x-tile (16 rows x 512)
    __shared__ float redD[WAVES * 16];
    __shared__ int   redI[WAVES * 16];
    __shared__ int   rowWin[16];

    const int tid   = threadIdx.x;
    const int wave  = tid >> 5;
    const int lane  = tid & 31;
    const int l15   = lane & 15;
    const int khalf = lane >> 4;        // which 16-lane half of the wave
    const int rowBase = blockIdx.x * 16;

    // Stage x tile into LDS (coalesced float4).
    {
        const float4* src = (const float4*)(x + (size_t)rowBase * Cd);
        float4* dst = (float4*)xs;
        for (int i = tid; i < (16 * Cd) / 4; i += NTHR) dst[i] = src[i];
    }
    __syncthreads();

    // A-fragment base: lane needs xs[row=l15][c + 2*khalf .. +1] each step.
    const float* aptr = xs + l15 * Cd + 2 * khalf;

    // Per-lane running argmin: lane holds 8 rows (M = r + 8*khalf) for the
    // single code column this lane owns in each k-iteration.
    float minD[8];
    int   minI[8];
#pragma unroll
    for (int r = 0; r < 8; ++r) { minD[r] = 3.4e38f; minI[r] = 0; }

    for (int k0 = 0; k0 < Kd; k0 += KTILE) {
        const int code = k0 + wave * 16 + l15;          // this lane's code column
        // B-fragment base: lane needs w[code][c + 2*khalf .. +1] each step.
        const float* bptr = weight + (size_t)code * Cd + 2 * khalf;

        if (k0 + KTILE < Kd)
            __builtin_prefetch(weight + (size_t)(code + KTILE) * Cd, 0, 1);

        const float wsqv = wsq[code];                   // hoisted ||w||^2

        v8f acc = {0.f, 0.f, 0.f, 0.f, 0.f, 0.f, 0.f, 0.f};

#pragma unroll 8
        for (int c = 0; c < Cd; c += 4) {
            v2f a = *(const v2f*)(aptr + c);            // ds_load_b64
            v2f b = *(const v2f*)(bptr + c);            // global_load_b64 (L2-resident)
            acc = __builtin_amdgcn_wmma_f32_16x16x4_f32(
                false, a, false, b, (short)0, acc, false, false);
        }

        // dist2 = x^2 + w^2 - 2*dot ; drop x^2 (constant per row).
#pragma unroll
        for (int r = 0; r < 8; ++r) {
            float d = wsqv - 2.0f * acc[r];
            if (d < minD[r] || (d == minD[r] && code < minI[r])) {
                minD[r] = d; minI[r] = code;
            }
        }
    }

    // Reduce argmin over the 16 code-lanes inside each half (xor stays in-half).
#pragma unroll
    for (int r = 0; r < 8; ++r) {
        float d = minD[r]; int i = minI[r];
#pragma unroll
        for (int off = 8; off >= 1; off >>= 1) {
            float od = __shfl_xor(d, off, 32);
            int   oi = __shfl_xor(i, off, 32);
            if (od < d || (od == d && oi < i)) { d = od; i = oi; }
        }
        minD[r] = d; minI[r] = i;
    }
    if (l15 == 0) {
#pragma unroll
        for (int r = 0; r < 8; ++r) {
            redD[wave * 16 + khalf * 8 + r] = minD[r];
            redI[wave * 16 + khalf * 8 + r] = minI[r];
        }
    }
    __syncthreads();

    // Final 4-wave reduction; emit index (+histogram).
    if (tid < 16) {
        float d = redD[tid]; int i = redI[tid];
        for (int w = 1; w < WAVES; ++w) {
            float od = redD[w * 16 + tid];
            int   oi = redI[w * 16 + tid];
            if (od < d || (od == d && oi < i)) { d = od; i = oi; }
        }
        rowWin[tid] = i;
        out_idx[rowBase + tid] = (float)i;
        atomicAdd(&hist[i], 1);
    }
    __syncthreads();

    // Gather quantized = weight[idx] for the 16 rows (coalesced float4).
    {
        float4* dst = (float4*)(out_q + (size_t)rowBase * Cd);
        for (int i = tid; i < (16 * Cd) / 4; i += NTHR) {
            int row = i >> 7;           // Cd/4 = 128 float4 per row
            int col = i & 127;
            const float4* src = (const float4*)(weight + (size_t)rowWin[row] * Cd);
            dst[i] = src[col];
        }
    }
}

// ||w_k||^2 for all K codes: one wave per code row, coalesced strided reads.
__global__ __launch_bounds__(256)
void vq_wsq_kernel(const float* __restrict__ weight, float* __restrict__ wsq)
{
    const int wave = threadIdx.x >> 5;
    const int lane = threadIdx.x & 31;
    const int row  = blockIdx.x * 8 + wave;     // 8 waves per block
    const float* wrow = weight + (size_t)row * Cd;

    float s = 0.f;
#pragma unroll
    for (int j = 0; j < Cd / 32; ++j) {
        float v = wrow[lane + 32 * j];
        s += v * v;
    }
#pragma unroll
    for (int off = 16; off >= 1; off >>= 1)
        s += __shfl_xor(s, off, 32);
    if (lane == 0) wsq[row] = s;
}

__global__ void vq_zero_hist(int* __restrict__ hist)
{
    hist[blockIdx.x * 256 + threadIdx.x] = 0;
}

__global__ void vq_usage_kernel(const int* __restrict__ hist, float* __restrict__ out)
{
    __shared__ int s[256];
    int zc = 0;
    for (int i = threadIdx.x; i < Kd; i += 256) zc += (hist[i] == 0) ? 1 : 0;
    s[threadIdx.x] = zc;
    __syncthreads();
    for (int off = 128; off > 0; off >>= 1) {
        if (threadIdx.x < off) s[threadIdx.x] += s[threadIdx.x + off];
        __syncthreads();
    }
    if (threadIdx.x == 0) out[0] = (float)s[0] / (float)Kd;
}

extern "C" void kernel_launch(void* const* d_in, const int* in_sizes, int n_in,
                              void* d_out, int out_size, void* d_ws, size_t ws_size,
                              hipStream_t stream)
{
    const float* x = (const float*)d_in[0];       // [B,H,W,C] f32
    const float* w = (const float*)d_in[1];       // [K,C]     f32

    float* out       = (float*)d_out;
    float* out_q     = out;                                    // [B*N, C]
    float* out_idx   = out + (size_t)NROW * Cd;                // [B*N]
    float* out_usage = out + (size_t)NROW * Cd + NROW;         // scalar

    // d_ws layout: [0, 16KB) hist (K ints), [16KB, 32KB) wsq (K floats)
    int*   hist = (int*)d_ws;
    float* wsq  = (float*)((char*)d_ws + (size_t)Kd * sizeof(int));

    vq_zero_hist<<<Kd / 256, 256, 0, stream>>>(hist);
    vq_wsq_kernel<<<Kd / 8, 256, 0, stream>>>(w, wsq);
    vq_argmin_kernel<<<NROW / 16, NTHR, 0, stream>>>(x, w, wsq, out_q, out_idx, hist);
    vq_usage_kernel<<<1, 256, 0, stream>>>(hist, out_usage);
}